// GCNEncoder_65704409694292
// MI455X (gfx1250) — compile-verified
//
#include <hip/hip_runtime.h>

typedef __attribute__((ext_vector_type(2))) float v2f;
typedef __attribute__((ext_vector_type(8))) float v8f;

// ---------------------------------------------------------------------------
// Degree / normalization:  deg = 1 (self loop) + #incoming edges; dinv = rsqrt
// ---------------------------------------------------------------------------
__global__ void k_deg_init(float* __restrict__ deg, int N) {
    int i = blockIdx.x * blockDim.x + threadIdx.x;
    if (i < N) deg[i] = 1.0f;   // self-loop contribution
}

__global__ void k_deg_acc(const long long* __restrict__ dst, float* __restrict__ deg, int E) {
    int e = blockIdx.x * blockDim.x + threadIdx.x;
    if (e < E) atomicAdd(&deg[(int)dst[e]], 1.0f);
}

__global__ void k_deg_rsqrt(float* __restrict__ deg, int N) {
    int i = blockIdx.x * blockDim.x + threadIdx.x;
    if (i < N) deg[i] = rsqrtf(deg[i]);   // deg >= 1 always (self loop)
}

// ---------------------------------------------------------------------------
// Weight transpose:  Wt[DOUT x DIN] = W[DIN x DOUT]^T   (tiny, once per call)
// Makes the WMMA B-fragment a single contiguous v2f load per K-step.
// ---------------------------------------------------------------------------
__global__ void k_transpose(const float* __restrict__ W, float* __restrict__ Wt,
                            int din, int dout) {
    int t = blockIdx.x * blockDim.x + threadIdx.x;
    if (t >= din * dout) return;
    int r = t / dout;   // row in W (K index)
    int c = t % dout;   // col in W (N index)
    Wt[(size_t)c * din + r] = W[(size_t)r * dout + c];
}

// ---------------------------------------------------------------------------
// Dense GEMM  H[N x DOUT] = X[N x DIN] @ W[DIN x DOUT] via V_WMMA_F32_16X16X4_F32
// One wave computes a full 16 x DOUT strip: one A fragment per K-step feeds
// DOUT/16 WMMAs, so X is read exactly once. W is pre-transposed so both A and
// B fragments are single 8B v2f loads. All dims multiples of 16 -> EXEC all-1s.
//
// Per K-step, ALL NT B-fragments are loaded into a register array first, then
// the NT WMMAs are issued — this gives the scheduler staging registers to
// clause the loads and overlap the next step's loads with current WMMAs
// (instead of a single staging pair forcing s_wait_loadcnt 0 per WMMA).
//
// A frag (16x4 f32): lane lo = row M; half hi selects K {0,1} vs {2,3}.
// B frag (4x16 f32): lane lo = col N; half hi selects K {0,1} vs {2,3}
//                    (from Wt[N][K], symmetric with A).
// D: VGPR v holds row (v + 8*hi), column lo.
// ---------------------------------------------------------------------------
template <int DIN, int DOUT>
__global__ void k_gemm_wmma(const float* __restrict__ X, const float* __restrict__ Wt,
                            float* __restrict__ H) {
    constexpr int NT = DOUT / 16;   // column tiles per wave

    const int lane = threadIdx.x & 31;
    const int lo = lane & 15;
    const int hi = lane >> 4;

    const size_t m0   = (size_t)blockIdx.x * 16;
    const float* xrow = X + (m0 + (size_t)lo) * DIN + 2 * hi;  // this lane's A row
    const float* wcol = Wt + (size_t)lo * DIN + 2 * hi;        // this lane's B col (tile 0)

    v8f acc[NT];
#pragma unroll
    for (int t = 0; t < NT; ++t)
        acc[t] = (v8f){0.f, 0.f, 0.f, 0.f, 0.f, 0.f, 0.f, 0.f};

#pragma unroll
    for (int k = 0; k < DIN; k += 4) {
        v2f a = *(const v2f*)(xrow + k);
        v2f b[NT];
#pragma unroll
        for (int t = 0; t < NT; ++t)
            b[t] = *(const v2f*)(wcol + (size_t)t * 16 * DIN + k);
#pragma unroll
        for (int t = 0; t < NT; ++t)
            acc[t] = __builtin_amdgcn_wmma_f32_16x16x4_f32(
                /*neg_a=*/false, a, /*neg_b=*/false, b[t],
                /*c_mod=*/(short)0, acc[t], /*reuse_a=*/false, /*reuse_b=*/false);
    }

#pragma unroll
    for (int t = 0; t < NT; ++t)
#pragma unroll
        for (int v = 0; v < 8; ++v)
            H[(m0 + (size_t)(v + 8 * hi)) * DOUT + t * 16 + lo] = acc[t][v];
}

// ---------------------------------------------------------------------------
// agg[i,:] = h[i,:] * dinv[i]^2 (self loop) + bias   (fully overwrites output)
// ---------------------------------------------------------------------------
template <int DOUT>
__global__ void k_init_agg(const float* __restrict__ H, const float* __restrict__ dinv,
                           const float* __restrict__ bias, float* __restrict__ A, int N) {
    constexpr int CPE = DOUT / 4;                 // float4 chunks per row
    long t = (long)blockIdx.x * blockDim.x + threadIdx.x;
    long total = (long)N * CPE;
    if (t >= total) return;
    int i = (int)(t / CPE);
    int c = (int)(t % CPE);
    float di = dinv[i];
    float s = di * di;
    float4 h = ((const float4*)H)[(size_t)i * CPE + c];
    float4 o;
    o.x = h.x * s + bias[4 * c + 0];
    o.y = h.y * s + bias[4 * c + 1];
    o.z = h.z * s + bias[4 * c + 2];
    o.w = h.w * s + bias[4 * c + 3];
    ((float4*)A)[(size_t)i * CPE + c] = o;
}

// ---------------------------------------------------------------------------
// Edge scatter: agg[dst,:] += h[src,:] * dinv[src]*dinv[dst]
// One thread per (edge, float4 chunk); gathers hit L2 (h fits in 192MB L2).
// ---------------------------------------------------------------------------
template <int DOUT>
__global__ void k_scatter(const float* __restrict__ H, const long long* __restrict__ eidx,
                          const float* __restrict__ dinv, float* __restrict__ A, int E) {
    constexpr int CPE = DOUT / 4;
    long t = (long)blockIdx.x * blockDim.x + threadIdx.x;
    long total = (long)E * CPE;
    if (t >= total) return;
    int e = (int)(t / CPE);
    int c = (int)(t % CPE);
    int s = (int)eidx[e];
    int d = (int)eidx[(long)E + e];
    float w = dinv[s] * dinv[d];
    float4 h = ((const float4*)H)[(size_t)s * CPE + c];
    float* o = A + (size_t)d * DOUT + 4 * c;
    atomicAdd(o + 0, h.x * w);
    atomicAdd(o + 1, h.y * w);
    atomicAdd(o + 2, h.z * w);
    atomicAdd(o + 3, h.w * w);
}

// ---------------------------------------------------------------------------
// In-place ReLU (float4)
// ---------------------------------------------------------------------------
__global__ void k_relu(float* __restrict__ A, long total4) {
    long t = (long)blockIdx.x * blockDim.x + threadIdx.x;
    if (t >= total4) return;
    float4 v = ((const float4*)A)[t];
    v.x = v.x > 0.f ? v.x : 0.f;
    v.y = v.y > 0.f ? v.y : 0.f;
    v.z = v.z > 0.f ? v.z : 0.f;
    v.w = v.w > 0.f ? v.w : 0.f;
    ((float4*)A)[t] = v;
}

// ---------------------------------------------------------------------------
// Host launch
// Inputs: x[N,128], edge_index[2,E] (int64), W1[128,128], b1[128],
//         W2[128,64], b2[64], W3[64,32], b3[32]
// ---------------------------------------------------------------------------
extern "C" void kernel_launch(void* const* d_in, const int* in_sizes, int n_in,
                              void* d_out, int out_size, void* d_ws, size_t ws_size,
                              hipStream_t stream) {
    constexpr int D_IN = 128, D_H1 = 128, D_H2 = 64, D_OUT = 32;

    const float*     x    = (const float*)d_in[0];
    const long long* eidx = (const long long*)d_in[1];
    const float*     W1   = (const float*)d_in[2];
    const float*     b1   = (const float*)d_in[3];
    const float*     W2   = (const float*)d_in[4];
    const float*     b2   = (const float*)d_in[5];
    const float*     W3   = (const float*)d_in[6];
    const float*     b3   = (const float*)d_in[7];
    float*           out  = (float*)d_out;

    const int N = in_sizes[0] / D_IN;
    const int E = in_sizes[1] / 2;

    // Workspace: dinv[N] | Wt1 | Wt2 | Wt3 | bufH[N*128] | bufA[N*128]
    char* wsb = (char*)d_ws;
    float* dinv = (float*)wsb;
    size_t off = ((size_t)N * sizeof(float) + 255) & ~(size_t)255;
    float* Wt1 = (float*)(wsb + off); off += (size_t)D_IN * D_H1 * sizeof(float);
    float* Wt2 = (float*)(wsb + off); off += (size_t)D_H1 * D_H2 * sizeof(float);
    float* Wt3 = (float*)(wsb + off); off += (size_t)D_H2 * D_OUT * sizeof(float);
    off = (off + 255) & ~(size_t)255;
    float* bufH = (float*)(wsb + off); off += (size_t)N * 128 * sizeof(float);
    float* bufA = (float*)(wsb + off);

    const int TB = 256;
    auto blocks = [](long total, int tb) { return (int)((total + tb - 1) / tb); };

    // --- normalization + weight transposes (independent, tiny) ---
    k_deg_init<<<blocks(N, TB), TB, 0, stream>>>(dinv, N);
    k_deg_acc<<<blocks(E, TB), TB, 0, stream>>>(eidx + E, dinv, E);  // dst row
    k_deg_rsqrt<<<blocks(N, TB), TB, 0, stream>>>(dinv, N);
    k_transpose<<<blocks(D_IN * D_H1, TB), TB, 0, stream>>>(W1, Wt1, D_IN, D_H1);
    k_transpose<<<blocks(D_H1 * D_H2, TB), TB, 0, stream>>>(W2, Wt2, D_H1, D_H2);
    k_transpose<<<blocks(D_H2 * D_OUT, TB), TB, 0, stream>>>(W3, Wt3, D_H2, D_OUT);

    // --- layer 1: 128 -> 128, ReLU ---
    k_gemm_wmma<D_IN, D_H1><<<N / 16, 32, 0, stream>>>(x, Wt1, bufH);
    k_init_agg<D_H1><<<blocks((long)N * (D_H1 / 4), TB), TB, 0, stream>>>(bufH, dinv, b1, bufA, N);
    k_scatter<D_H1><<<blocks((long)E * (D_H1 / 4), TB), TB, 0, stream>>>(bufH, eidx, dinv, bufA, E);
    k_relu<<<blocks((long)N * (D_H1 / 4), TB), TB, 0, stream>>>(bufA, (long)N * (D_H1 / 4));

    // --- layer 2: 128 -> 64, ReLU ---
    k_gemm_wmma<D_H1, D_H2><<<N / 16, 32, 0, stream>>>(bufA, Wt2, bufH);
    k_init_agg<D_H2><<<blocks((long)N * (D_H2 / 4), TB), TB, 0, stream>>>(bufH, dinv, b2, bufA, N);
    k_scatter<D_H2><<<blocks((long)E * (D_H2 / 4), TB), TB, 0, stream>>>(bufH, eidx, dinv, bufA, E);
    k_relu<<<blocks((long)N * (D_H2 / 4), TB), TB, 0, stream>>>(bufA, (long)N * (D_H2 / 4));

    // --- layer 3: 64 -> 32, no activation, straight into d_out ---
    k_gemm_wmma<D_H2, D_OUT><<<N / 16, 32, 0, stream>>>(bufA, Wt3, bufH);
    k_init_agg<D_OUT><<<blocks((long)N * (D_OUT / 4), TB), TB, 0, stream>>>(bufH, dinv, b3, out, N);
    k_scatter<D_OUT><<<blocks((long)E * (D_OUT / 4), TB), TB, 0, stream>>>(bufH, eidx, dinv, out, E);
}